// path_generator_BS_OOSP_33603824123993
// MI455X (gfx1250) — compile-verified
//
#include <hip/hip_runtime.h>
#include <hip/hip_bf16.h>

// GBM path generation on MI455X (gfx1250). Bandwidth-bound: 124 MB in +
// 124 MB out => ~10.6 us floor at 23.3 TB/s. The per-path inclusive prefix
// sum is computed on the matrix pipe as P = R x U (U upper-triangular ones)
// with 12 chained V_WMMA_F32_16X16X4_F32 per 16-path tile; the VALU only
// does staging addresses, 4 triangular B pairs, and fma+v_exp epilogue.

typedef __attribute__((ext_vector_type(2))) float v2f;
typedef __attribute__((ext_vector_type(8))) float v8f;

#define TSTEPS 31
#define TILE_PATHS 16
#define WAVES_PER_BLOCK 8
#define DT (1.0f / 30.0f)

__global__ __launch_bounds__(256) void gbm_wmma_scan_kernel(
    const float* __restrict__ sigma,
    const float* __restrict__ Z,
    float* __restrict__ out,
    int n_tiles, int tiles_per_inst)
{
    // flat [16 x 31] tile per wave, padded to 512 floats for b128 alignment
    __shared__ __align__(16) float lds[WAVES_PER_BLOCK][512];

    const int lane = threadIdx.x & 31;
    const int wv   = threadIdx.x >> 5;
    const int tile = blockIdx.x * WAVES_PER_BLOCK + wv;
    if (tile >= n_tiles) return;            // whole-wave uniform exit

    float* tp = lds[wv];
    const int base = tile * (TILE_PATHS * TSTEPS);   // 496 floats, 16B-aligned start

    // ---- Stage 124 float4 chunks -> LDS, b128 loads + b128 LDS stores ----
    const float4* __restrict__ zsrc4 = (const float4*)(Z + base);
    float4* lp = (float4*)tp;
#pragma unroll
    for (int j = 0; j < 3; ++j) {
        int chunk = j * 32 + lane;          // 0..95
        lp[chunk] = zsrc4[chunk];
    }
    if (lane < 28) {                        // tail chunks 96..123
        int chunk = 96 + lane;
        lp[chunk] = zsrc4[chunk];
    }

    __builtin_amdgcn_wave_barrier();
    asm volatile("s_wait_dscnt 0" ::: "memory");   // cross-lane DS RAW ordering

    // ---- WMMA prefix-sum: P[16x32] = R[16x32] x U[32x32], U[k,t]=(k<=t) ----
    const int inst = tile / tiles_per_inst;
    const float sig     = sigma[inst];
    const float driftdt = -0.5f * sig * sig * DT;  // MEAN == 0

    const int arow = lane & 15;             // A row M; also B column within tile
    const int koff = (lane >> 4) << 1;      // lanes 16-31 carry K+2,K+3 (ISA A layout)
    const int abase = arow * TSTEPS + koff; // flat stride-31: odd => bank-conflict-free

    // Triangular B pairs for cols [0,16); reused (shifted) for cols [16,32).
    const v2f ones = {1.0f, 1.0f};
    v2f bt[4];
#pragma unroll
    for (int s = 0; s < 4; ++s) {
        int rx = 4 * s + koff;
        bt[s].x = (rx     <= arow) ? 1.0f : 0.0f;
        bt[s].y = (rx + 1 <= arow) ? 1.0f : 0.0f;
    }

    v8f c0 = {};   // output cols t = 0..15
    v8f c1 = {};   // output cols t = 16..31
#pragma unroll
    for (int s = 0; s < 8; ++s) {
        v2f a;
        a.x = tp[abase + 4 * s + 0];
        a.y = tp[abase + 4 * s + 1];
        if (s == 0 && lane < 16)  a.x = 0.0f;   // K=0: Z[...,0] unused by reference
        if (s == 7 && lane >= 16) a.y = 0.0f;   // K=31: pad column

        if (s < 4) {
            // cols [0,16): triangular; cols [16,32): K<16 <= t+16 -> all-ones
            c0 = __builtin_amdgcn_wmma_f32_16x16x4_f32(
                     false, a, false, bt[s], (short)0, c0, false, false);
            c1 = __builtin_amdgcn_wmma_f32_16x16x4_f32(
                     false, a, false, ones, (short)0, c1, false, false);
        } else {
            // cols [16,32), K in [16,32): (K <= t+16) == (K-16 <= t) -> bt[s-4]
            c1 = __builtin_amdgcn_wmma_f32_16x16x4_f32(
                     false, a, false, bt[s - 4], (short)0, c1, false, false);
        }
    }

    // ---- Epilogue: S = exp2(fma(sig*log2e, P, t*driftdt*log2e + log2(100))) ----
    const float LOG2E    = 1.4426950408889634f;
    const float LOG2_100 = 6.643856189774725f;
    const float sl  = sig * LOG2E;
    const float dl  = driftdt * LOG2E;
    const int   t0  = lane & 15;
    const float ct0 = (float)t0 * dl + LOG2_100;
    const float ct1 = (float)(t0 + 16) * dl + LOG2_100;

    float* __restrict__ obase = out + base;
    const int prow0 = (lane >> 4) * 8;      // C layout: vgpr v -> row v (+8 hi half)
#pragma unroll
    for (int v = 0; v < 8; ++v) {
        float* orow = obase + (prow0 + v) * TSTEPS;
        orow[t0] = __builtin_amdgcn_exp2f(__builtin_fmaf(sl, c0[v], ct0));
        if (t0 != 15) {                     // t = 31 does not exist
            orow[t0 + 16] = __builtin_amdgcn_exp2f(__builtin_fmaf(sl, c1[v], ct1));
        }
    }
}

extern "C" void kernel_launch(void* const* d_in, const int* in_sizes, int n_in,
                              void* d_out, int out_size, void* d_ws, size_t ws_size,
                              hipStream_t stream) {
    const float* sigma = (const float*)d_in[0];
    const float* Z     = (const float*)d_in[1];
    float* out         = (float*)d_out;

    const int n_inst = in_sizes[0];                            // 100
    const int paths  = in_sizes[1] / TSTEPS;                   // 1,000,000
    const int n_tiles = paths / TILE_PATHS;                    // 62,500
    const int tiles_per_inst = (paths / n_inst) / TILE_PATHS;  // 625

    dim3 block(32 * WAVES_PER_BLOCK);
    dim3 grid((n_tiles + WAVES_PER_BLOCK - 1) / WAVES_PER_BLOCK);
    gbm_wmma_scan_kernel<<<grid, block, 0, stream>>>(sigma, Z, out,
                                                     n_tiles, tiles_per_inst);
}